// TimeLayerCrosscoder_90984587198484
// MI455X (gfx1250) — compile-verified
//
#include <hip/hip_runtime.h>

typedef __attribute__((ext_vector_type(2))) float v2f;
typedef __attribute__((ext_vector_type(8))) float v8f;
typedef __attribute__((ext_vector_type(4))) unsigned int u32x4;
typedef __attribute__((ext_vector_type(8))) int i32x8;
typedef __attribute__((ext_vector_type(4))) int i32x4;

#define D_IN   2304
#define D_SAE  2048
#define TT     4
#define LLAY   6
#define TLNUM  24            // TT*LLAY
#define KTOP   128
#define BB     32
#define NLAT   (TLNUM * D_SAE)   // 49152 latents per batch row

// ---------------------------------------------------------------------------
// TDM helper: issue a 2D fp32 tile DMA Global->LDS (CDNA5 ISA).
// Tile = tile_y rows x tile_x elements, global row stride = stride0 elements.
// Hardware padding: 4 dwords inserted every 64 dwords -> LDS pitch 68 floats
// (pad_interval code 5 = 64 dw, pad_amount code 3 = 4 dw).
// Uses the 6-arg builtin form (clang-23 / therock-10.0 headers).
// ---------------------------------------------------------------------------
__device__ __forceinline__ void tdm_load_tile_2d(
    unsigned lds_byte_addr, const void* gptr,
    unsigned tile_x, unsigned tile_y, unsigned long long stride0_elems)
{
    const unsigned long long ga = (unsigned long long)gptr;
    u32x4 g0;
    g0[0] = 1u;                                        // count=1 (valid user D#)
    g0[1] = lds_byte_addr;                             // lds_addr
    g0[2] = (unsigned)ga;                              // global_addr[31:0]
    g0[3] = (unsigned)((ga >> 32) & 0x01FFFFFFu)       // global_addr[56:32]
          | (2u << 30);                                // type=2 ("image")
    i32x8 g1;
    g1[0] = (int)((2u << 16)                           // data_size = 4B
                | (1u << 20)                           // pad_enable
                | (5u << 22)                           // pad_interval: 64 dwords
                | (3u << 25));                         // pad_amount: 4 dwords
    g1[1] = (int)((tile_x & 0xFFFFu) << 16);           // tensor_dim0[15:0] (=tile_x)
    g1[2] = (int)(((tile_x >> 16) & 0xFFFFu)
                | ((tile_y & 0xFFFFu) << 16));         // tensor_dim0 hi | tensor_dim1 lo
    g1[3] = (int)(((tile_y >> 16) & 0xFFFFu)
                | ((tile_x & 0xFFFFu) << 16));         // tensor_dim1 hi | tile_dim0
    g1[4] = (int)(tile_y & 0xFFFFu);                   // tile_dim1 (tile_dim2 = 0)
    g1[5] = (int)(unsigned)(stride0_elems & 0xFFFFFFFFull);  // dim0_stride[31:0]
    g1[6] = (int)(unsigned)((stride0_elems >> 32) & 0xFFFFull); // dim0_stride[47:32]
    g1[7] = 0;                                         // dim1_stride (unused for 2D)
    const i32x4 gz4 = {0, 0, 0, 0};                    // groups 2/3: 2D tensor
    const i32x8 gz8 = {0, 0, 0, 0, 0, 0, 0, 0};        // extra descriptor words: unused
    __builtin_amdgcn_tensor_load_to_lds(g0, g1, gz4, gz4, gz8, 0);
}

// ---------------------------------------------------------------------------
// Kernel 1: encode  pre[b, tl*D_SAE + s] = sum_d x[b,tl,d]*W_enc[tl,d,s] + b_enc[s]
// Grid (TLNUM, D_SAE/64), 256 threads = 8 wave32s. Each wave owns one 16x16
// output tile of the 32x64 block tile, accumulating with V_WMMA_F32_16X16X4_F32.
// Wave 0 drives a double-buffered TDM pipeline (Global->LDS, TENSORcnt).
// ---------------------------------------------------------------------------
__global__ __launch_bounds__(256) void cc_encode_wmma(
    const float* __restrict__ x, const float* __restrict__ W_enc,
    const float* __restrict__ b_enc, float* __restrict__ pre)
{
    constexpr int KC = 64;                 // K-chunk staged in LDS
    constexpr int PITCH = KC + 4;          // 68 floats (TDM pad reproduces this)
    constexpr int NITER = D_IN / KC;       // 36
    __shared__ float Xs[2][32][PITCH];     // A operand: Xs[m][k], row-major
    __shared__ float Ws[2][KC][PITCH];     // B operand: Ws[k][n], row-major (TDM tile)

    const int tl  = blockIdx.x;            // 0..23
    const int n0  = blockIdx.y * 64;       // output column base in D_SAE
    const int tid = threadIdx.x;
    const int wave = tid >> 5, lane = tid & 31;
    const int m0 = (wave >> 2) << 4;       // 0 | 16  (batch rows)
    const int nw = (wave & 3) << 4;        // 0..48   (cols within block tile)

    const float* Xbase = x + (size_t)tl * D_IN;                   // row stride TL*D_IN
    const float* Wbase = W_enc + (size_t)tl * D_IN * D_SAE + n0;  // row stride D_SAE

    v8f acc = {};

    // Prologue: issue chunk 0 into buffer 0 (wave 0 only; TDM ignores EXEC).
    if (wave == 0) {
        tdm_load_tile_2d((unsigned)(size_t)&Xs[0][0][0], Xbase, KC, 32,
                         (unsigned long long)(TLNUM * D_IN));
        tdm_load_tile_2d((unsigned)(size_t)&Ws[0][0][0], Wbase, 64, KC,
                         (unsigned long long)D_SAE);
    }

    const int mr = m0 + (lane & 15);
    const int nr = nw + (lane & 15);
    const int kh = (lane >> 4) << 1;       // 0 or 2 (A/B K-halves per §7.12.2)

    for (int it = 0; it < NITER; ++it) {
        const int buf = it & 1;
        if (wave == 0) {
            if (it + 1 < NITER) {
                const int kc = (it + 1) * KC;
                tdm_load_tile_2d((unsigned)(size_t)&Xs[buf ^ 1][0][0],
                                 Xbase + kc, KC, 32,
                                 (unsigned long long)(TLNUM * D_IN));
                tdm_load_tile_2d((unsigned)(size_t)&Ws[buf ^ 1][0][0],
                                 Wbase + (size_t)kc * D_SAE, 64, KC,
                                 (unsigned long long)D_SAE);
                __builtin_amdgcn_s_wait_tensorcnt(2);   // current pair resident
            } else {
                __builtin_amdgcn_s_wait_tensorcnt(0);   // last pair resident
            }
        }
        __syncthreads();   // publish TDM-landed tiles to all waves

        const float (*Xb)[PITCH] = Xs[buf];
        const float (*Wb)[PITCH] = Ws[buf];
        #pragma unroll
        for (int k = 0; k < KC; k += 4) {
            // A 16x4: lanes 0-15 hold {K,K+1} of row M=lane; lanes 16-31 {K+2,K+3}
            v2f a = *(const v2f*)&Xb[mr][k + kh];
            // B 4x16: lane n holds {B[K][n], B[K+1][n]} (rows k+kh, k+kh+1)
            v2f b;
            b.x = Wb[k + kh + 0][nr];
            b.y = Wb[k + kh + 1][nr];
            acc = __builtin_amdgcn_wmma_f32_16x16x4_f32(
                /*neg_a=*/false, a, /*neg_b=*/false, b,
                /*c_mod=*/(short)0, acc, /*reuse_a=*/false, /*reuse_b=*/false);
        }
        __syncthreads();   // all reads of buf done before it is re-issued
    }

    // Epilogue: C/D layout — VGPR r: lanes 0-15 -> M=r, lanes 16-31 -> M=r+8; N=lane&15
    const int n     = nw + (lane & 15);
    const float be  = b_enc[n0 + n];
    const int mbase = m0 + ((lane >> 4) << 3);
    #pragma unroll
    for (int r = 0; r < 8; ++r) {
        const int m = mbase + r;   // batch row
        pre[(size_t)m * NLAT + (size_t)tl * D_SAE + n0 + n] = acc[r] + be;
    }
}

// ---------------------------------------------------------------------------
// Kernel 2: exact global top-K per batch row via 4-pass 8-bit radix select on
// the order-preserving uint mapping of f32. Zeroes z, writes z=relu(v) at the
// K winners, and emits a compact (idx,val) list for the sparse decoder.
// pre is 6.3 MB -> fully L2-resident, so the extra passes are cheap.
// ---------------------------------------------------------------------------
__global__ __launch_bounds__(256) void cc_topk(
    const float* __restrict__ pre, float* __restrict__ z,
    int* __restrict__ lidx, float* __restrict__ lval, float* __restrict__ loss)
{
    __shared__ unsigned hist[256];
    __shared__ int s_bucket, s_krem;
    __shared__ unsigned s_eq, s_cnt;

    const int b = blockIdx.x, tid = threadIdx.x;
    const float* row = pre + (size_t)b * NLAT;
    float* zrow = z + (size_t)b * NLAT;
    if (b == 0 && tid == 0) loss[0] = 0.0f;   // decode kernel accumulates after us

    unsigned prefix = 0u, pmask = 0u;
    int krem = KTOP;
    for (int shift = 24; shift >= 0; shift -= 8) {
        hist[tid] = 0u;
        __syncthreads();
        for (int i = tid; i < NLAT; i += 256) {
            if (shift == 24) zrow[i] = 0.0f;   // fold z-clear into first scan
            unsigned u = __float_as_uint(row[i]);
            u = (u & 0x80000000u) ? ~u : (u | 0x80000000u);
            if ((u & pmask) == prefix) atomicAdd(&hist[(u >> shift) & 0xFFu], 1u);
        }
        __syncthreads();
        if (tid == 0) {
            int acc = 0, bk = 255;
            for (; bk >= 0; --bk) {
                const int h = (int)hist[bk];
                if (acc + h >= krem) break;
                acc += h;
            }
            if (bk < 0) bk = 0;
            s_bucket = bk;
            s_krem = krem - acc;
        }
        __syncthreads();
        prefix |= ((unsigned)s_bucket) << shift;
        pmask  |= 0xFFu << shift;
        krem = s_krem;
        __syncthreads();
    }
    if (tid == 0) { s_eq = 0u; s_cnt = 0u; }
    __syncthreads();

    const unsigned thr = prefix;   // exact K-th value's key
    for (int i = tid; i < NLAT; i += 256) {
        const float v = row[i];
        unsigned u = __float_as_uint(v);
        u = (u & 0x80000000u) ? ~u : (u | 0x80000000u);
        bool take = (u > thr);
        if (!take && u == thr) take = (atomicAdd(&s_eq, 1u) < (unsigned)krem);
        if (take) {
            const unsigned slot = atomicAdd(&s_cnt, 1u);   // exactly KTOP taken
            const float zv = v > 0.0f ? v : 0.0f;          // relu
            lidx[b * KTOP + slot] = i;
            lval[b * KTOP + slot] = zv;
            zrow[i] = zv;
        }
    }
}

// ---------------------------------------------------------------------------
// Kernel 3: sparse decode + loss. One block per (b, tl): x_hat = b_dec +
// sum over active latents of this (tl): val * W_dec[tl, s, :]; then squared
// error reduced and atomically folded into the mean loss. Exploiting the
// K=128 sparsity keeps W_dec traffic at ~38 MB instead of 453 MB dense.
// ---------------------------------------------------------------------------
__global__ __launch_bounds__(256) void cc_decode(
    const float* __restrict__ x, const float* __restrict__ W_dec,
    const float* __restrict__ b_dec, const int* __restrict__ lidx,
    const float* __restrict__ lval, float* __restrict__ x_hat,
    float* __restrict__ loss)
{
    __shared__ int   sIdx[KTOP];
    __shared__ float sVal[KTOP];
    __shared__ float red[256];

    const int blk = blockIdx.x;
    const int b = blk / TLNUM, tl = blk % TLNUM;
    const int tid = threadIdx.x;

    if (tid < KTOP) {
        sIdx[tid] = lidx[b * KTOP + tid];
        sVal[tid] = lval[b * KTOP + tid];
    }
    __syncthreads();

    float acc[9];                                   // D_IN = 9 * 256
    #pragma unroll
    for (int j = 0; j < 9; ++j) acc[j] = b_dec[(size_t)tl * D_IN + tid + j * 256];

    for (int e = 0; e < KTOP; ++e) {                // uniform across block
        const int fi  = sIdx[e];
        const int etl = fi >> 11;                   // / D_SAE
        if (etl == tl) {
            const float v = sVal[e];
            const float* wr = W_dec + ((size_t)tl * D_SAE + (fi & (D_SAE - 1))) * D_IN;
            #pragma unroll
            for (int j = 0; j < 9; ++j) acc[j] += v * wr[tid + j * 256];
        }
    }

    const float* xr = x + ((size_t)b * TLNUM + tl) * D_IN;
    float* xh = x_hat + ((size_t)b * TLNUM + tl) * D_IN;
    float ss = 0.0f;
    #pragma unroll
    for (int j = 0; j < 9; ++j) {
        const int d = tid + j * 256;
        const float v = acc[j];
        xh[d] = v;
        const float df = v - xr[d];
        ss += df * df;
    }
    red[tid] = ss;
    __syncthreads();
    for (int off = 128; off > 0; off >>= 1) {
        if (tid < off) red[tid] += red[tid + off];
        __syncthreads();
    }
    if (tid == 0) atomicAdd(loss, red[0] * (1.0f / (float)(BB * TLNUM)));
}

// ---------------------------------------------------------------------------
extern "C" void kernel_launch(void* const* d_in, const int* in_sizes, int n_in,
                              void* d_out, int out_size, void* d_ws, size_t ws_size,
                              hipStream_t stream)
{
    const float* x     = (const float*)d_in[0];
    const float* W_enc = (const float*)d_in[1];
    const float* W_dec = (const float*)d_in[2];
    const float* b_enc = (const float*)d_in[3];
    const float* b_dec = (const float*)d_in[4];

    float* out   = (float*)d_out;
    float* loss  = out;                                      // [1]
    float* x_hat = out + 1;                                  // [B*TL*D_IN]
    float* z     = out + 1 + (size_t)BB * TLNUM * D_IN;      // [B*TL*D_SAE]

    float* pre  = (float*)d_ws;                              // [B*NLAT] = 6.3 MB
    int*   lidx = (int*)(pre + (size_t)BB * NLAT);           // [B*K]
    float* lval = (float*)(lidx + BB * KTOP);                // [B*K]

    dim3 egrid(TLNUM, D_SAE / 64);
    cc_encode_wmma<<<egrid, 256, 0, stream>>>(x, W_enc, b_enc, pre);
    cc_topk<<<BB, 256, 0, stream>>>(pre, z, lidx, lval, loss);
    cc_decode<<<BB * TLNUM, 256, 0, stream>>>(x, W_dec, b_dec, lidx, lval, x_hat, loss);
}